// SAGEConvWithEdgeAttrAndAttention_24051816857687
// MI455X (gfx1250) — compile-verified
//
#include <hip/hip_runtime.h>
#include <hip/hip_bf16.h>

typedef float v2f __attribute__((ext_vector_type(2)));
typedef float v8f __attribute__((ext_vector_type(8)));

#define N_NODES 100000
#define N_EDGES 1000000
#define IN_C 128
#define OUT_C 64
#define EDGE_D 32

__device__ __forceinline__ v8f wmma_f32_16x16x4(v2f a, v2f b, v8f c) {
    // 8 args: (neg_a, A, neg_b, B, c_mod, C, reuse_a, reuse_b)
    return __builtin_amdgcn_wmma_f32_16x16x4_f32(false, a, false, b, (short)0, c,
                                                 false, false);
}

// ---------------------------------------------------------------------------
// Kernel 1: node GEMMs  h_l = x @ W_l   and   xWr = x @ W_r
// One wave handles a 16-node tile; f32 WMMA 16x16x4, K=128 (32 steps), N=64
// (4 tiles), two weight matrices -> 8 accumulators.
// A layout (16x4 f32): v0 = K=(0|2) for lanes (0-15|16-31), v1 = K=(1|3).
// B layout (4x16 f32): v0 = row K=(0|2), v1 = row K=(1|3), N striped on lanes.
// C layout (16x16 f32): VGPR r: lanes0-15 -> M=r, lanes16-31 -> M=r+8.
// ---------------------------------------------------------------------------
__global__ void node_gemm_kernel(const float* __restrict__ x,
                                 const float* __restrict__ W_l,
                                 const float* __restrict__ W_r,
                                 float* __restrict__ h_l,
                                 float* __restrict__ xWr) {
    const int wave = (int)((blockIdx.x * blockDim.x + threadIdx.x) >> 5);
    if (wave >= N_NODES / 16) return;          // wave-uniform: EXEC stays full
    const int lane = threadIdx.x & 31;
    const int half = lane >> 4;
    const int l    = lane & 15;
    const int node0 = wave * 16;
    const float* xrow = x + (size_t)(node0 + l) * IN_C + 2 * half;

    v8f accL[4], accR[4];
#pragma unroll
    for (int nt = 0; nt < 4; ++nt) {
        accL[nt] = (v8f){0.f,0.f,0.f,0.f,0.f,0.f,0.f,0.f};
        accR[nt] = (v8f){0.f,0.f,0.f,0.f,0.f,0.f,0.f,0.f};
    }

    for (int kk = 0; kk < IN_C; kk += 4) {
        v2f a = *(const v2f*)(xrow + kk);       // 8B-aligned (even offsets)
        const int krow = kk + 2 * half;
#pragma unroll
        for (int nt = 0; nt < 4; ++nt) {
            const int n = nt * 16 + l;
            v2f bl, br;
            bl.x = W_l[(size_t)krow * OUT_C + n];
            bl.y = W_l[(size_t)(krow + 1) * OUT_C + n];
            br.x = W_r[(size_t)krow * OUT_C + n];
            br.y = W_r[(size_t)(krow + 1) * OUT_C + n];
            accL[nt] = wmma_f32_16x16x4(a, bl, accL[nt]);
            accR[nt] = wmma_f32_16x16x4(a, br, accR[nt]);
        }
    }

#pragma unroll
    for (int nt = 0; nt < 4; ++nt) {
#pragma unroll
        for (int r = 0; r < 8; ++r) {
            const int m = r + half * 8;
            const int n = nt * 16 + l;
            h_l[(size_t)(node0 + m) * OUT_C + n] = accL[nt][r];
            xWr[(size_t)(node0 + m) * OUT_C + n] = accR[nt][r];
        }
    }
}

// ---------------------------------------------------------------------------
// Kernel 2: scatter-add of h_l[row] into agg[col] + degree count.
// thread t -> (edge = t/64, channel = t%64): coalesced per-edge 256B bursts.
// ---------------------------------------------------------------------------
__global__ void scatter_agg_kernel(const int* __restrict__ ei,
                                   const float* __restrict__ h_l,
                                   float* __restrict__ agg,
                                   float* __restrict__ cnt) {
    const long long t = (long long)blockIdx.x * blockDim.x + threadIdx.x;
    const int e = (int)(t >> 6);
    const int c = (int)(t & 63);
    if (e >= N_EDGES) return;
    const int r = ei[e];                 // source (row)
    const int d = ei[N_EDGES + e];       // target (col)
    atomicAdd(&agg[(size_t)d * OUT_C + c], h_l[(size_t)r * OUT_C + c]);
    if (c == 0) atomicAdd(&cnt[d], 1.0f);
}

// ---------------------------------------------------------------------------
// Kernel 3: out_base = agg / max(cnt,1) + b_l + xWr  (in-place into xWr),
// and zero agg so it can be reused as the attention accumulator.
// ---------------------------------------------------------------------------
__global__ void node_finalize_kernel(const float* __restrict__ cnt,
                                     const float* __restrict__ b_l,
                                     float* __restrict__ agg,
                                     float* __restrict__ out_base) {
    const int i = blockIdx.x * blockDim.x + threadIdx.x;
    if (i >= N_NODES * OUT_C) return;
    const int n = i >> 6;
    const int c = i & 63;
    const float d = fmaxf(cnt[n], 1.0f);
    out_base[i] = agg[i] / d + b_l[c] + out_base[i];
    agg[i] = 0.0f;
}

// ---------------------------------------------------------------------------
// Kernel 4: edge pipeline. One wave = 16 edges.
//   e = edge_attr @ W_e + b_e           (32 f32 WMMAs, fully unrolled)
//   att = sigmoid(out_base[col].Wa1 + e.Wa2 + b_att)  (shfl_xor reductions)
//   accum[col] += att * e               (float atomics)
// Per-edge scalars are kept in lane-indexed registers and moved with __shfl,
// so no LDS ordering hazards.
// ---------------------------------------------------------------------------
__global__ void edge_kernel(const int* __restrict__ ei,
                            const float* __restrict__ edge_attr,
                            const float* __restrict__ W_e,
                            const float* __restrict__ b_e,
                            const float* __restrict__ W_att,
                            const float* __restrict__ b_att,
                            const float* __restrict__ out_base,
                            float* __restrict__ accum) {
    const int wave = (int)((blockIdx.x * blockDim.x + threadIdx.x) >> 5);
    if (wave >= N_EDGES / 16) return;          // wave-uniform
    const int lane = threadIdx.x & 31;
    const int half = lane >> 4;
    const int l    = lane & 15;
    const int e0   = wave * 16;
    const float* arow = edge_attr + (size_t)(e0 + l) * EDGE_D + 2 * half;

    v8f acc[4];
#pragma unroll
    for (int nt = 0; nt < 4; ++nt)
        acc[nt] = (v8f){0.f,0.f,0.f,0.f,0.f,0.f,0.f,0.f};

#pragma unroll
    for (int kk = 0; kk < EDGE_D; kk += 4) {
        v2f a = *(const v2f*)(arow + kk);
        const int krow = kk + 2 * half;
#pragma unroll
        for (int nt = 0; nt < 4; ++nt) {
            const int n = nt * 16 + l;
            v2f b;
            b.x = W_e[(size_t)krow * OUT_C + n];
            b.y = W_e[(size_t)(krow + 1) * OUT_C + n];
            acc[nt] = wmma_f32_16x16x4(a, b, acc[nt]);
        }
    }

    // bias + per-edge partial dot e . W_att[64:128]
    float be[4], wa2[4];
#pragma unroll
    for (int nt = 0; nt < 4; ++nt) {
        be[nt]  = b_e[nt * 16 + l];
        wa2[nt] = W_att[OUT_C + nt * 16 + l];
    }
    float pack = 0.f;   // lane j(<8) holds s_e[j]; lane 16+j holds s_e[8+j]
#pragma unroll
    for (int r = 0; r < 8; ++r) {
        float p = 0.f;
#pragma unroll
        for (int nt = 0; nt < 4; ++nt) {
            acc[nt][r] += be[nt];
            p += acc[nt][r] * wa2[nt];
        }
        p += __shfl_xor(p, 1, 32);   // reduce within 16-lane half
        p += __shfl_xor(p, 2, 32);
        p += __shfl_xor(p, 4, 32);
        p += __shfl_xor(p, 8, 32);
        if (l == r) pack = p;
    }

    // attention scalar per edge
    const float wa1_lo = W_att[lane];
    const float wa1_hi = W_att[32 + lane];
    const float batt   = b_att[0];
    float attk = 0.f;                 // lane m holds att of edge m
#pragma unroll
    for (int m = 0; m < 16; ++m) {
        const float se = __shfl(pack, (m & 7) + ((m >> 3) << 4), 32);
        const int cm = ei[N_EDGES + e0 + m];          // uniform -> scalar load
        const float* ob = out_base + (size_t)cm * OUT_C;
        float p = ob[lane] * wa1_lo + ob[32 + lane] * wa1_hi;
        p += __shfl_xor(p, 1, 32);
        p += __shfl_xor(p, 2, 32);
        p += __shfl_xor(p, 4, 32);
        p += __shfl_xor(p, 8, 32);
        p += __shfl_xor(p, 16, 32);
        const float s = p + se + batt;
        const float att = 1.0f / (1.0f + __expf(-s));
        if (lane == m) attk = att;
    }

    // gated scatter-add
#pragma unroll
    for (int r = 0; r < 8; ++r) {
        const int m  = r + half * 8;
        const int cm = ei[N_EDGES + e0 + m];
        const float attm = __shfl(attk, m, 32);
        float* dst = accum + (size_t)cm * OUT_C;
#pragma unroll
        for (int nt = 0; nt < 4; ++nt)
            atomicAdd(&dst[nt * 16 + l], attm * acc[nt][r]);
    }
}

// ---------------------------------------------------------------------------
// Kernel 5: per-channel sum / sumsq of (out_base + accum) via block partials.
// ---------------------------------------------------------------------------
__global__ void stats_kernel(const float* __restrict__ base,
                             const float* __restrict__ accum,
                             float* __restrict__ stats) {  // [0:64]=sum, [64:128]=sumsq
    const int t = threadIdx.x;
    const int c = t & 63;
    const int stride = (int)(gridDim.x * blockDim.x) >> 6;  // nodes per sweep
    float s = 0.f, s2 = 0.f;
    for (int n = blockIdx.x * (blockDim.x >> 6) + (t >> 6); n < N_NODES; n += stride) {
        const float v = base[(size_t)n * OUT_C + c] + accum[(size_t)n * OUT_C + c];
        s += v;
        s2 += v * v;
    }
    atomicAdd(&stats[c], s);
    atomicAdd(&stats[64 + c], s2);
}

// ---------------------------------------------------------------------------
// Kernel 6: batchnorm + residual-double + relu
// ---------------------------------------------------------------------------
__global__ void final_kernel(const float* __restrict__ base,
                             const float* __restrict__ accum,
                             const float* __restrict__ stats,
                             const float* __restrict__ gamma,
                             const float* __restrict__ beta,
                             float* __restrict__ out) {
    const int i = blockIdx.x * blockDim.x + threadIdx.x;
    if (i >= N_NODES * OUT_C) return;
    const int c = i & 63;
    const float inv_n = 1.0f / (float)N_NODES;
    const float v    = base[i] + accum[i];
    const float mean = stats[c] * inv_n;
    const float var  = stats[64 + c] * inv_n - mean * mean;
    const float y = gamma[c] * (v - mean) * rsqrtf(var + 1e-5f) + beta[c];
    const float z = y + y;
    out[i] = fmaxf(z, 0.0f);
}

// ---------------------------------------------------------------------------
extern "C" void kernel_launch(void* const* d_in, const int* in_sizes, int n_in,
                              void* d_out, int out_size, void* d_ws, size_t ws_size,
                              hipStream_t stream) {
    const float* x         = (const float*)d_in[0];
    const int*   ei        = (const int*)d_in[1];
    const float* edge_attr = (const float*)d_in[2];
    const float* W_l       = (const float*)d_in[3];
    const float* b_l       = (const float*)d_in[4];
    const float* W_r       = (const float*)d_in[5];
    const float* W_e       = (const float*)d_in[6];
    const float* b_e       = (const float*)d_in[7];
    const float* W_att     = (const float*)d_in[8];
    const float* b_att     = (const float*)d_in[9];
    const float* gamma     = (const float*)d_in[10];
    const float* beta      = (const float*)d_in[11];

    // workspace layout (floats)
    float* ws    = (float*)d_ws;
    float* h_l   = ws;                                   // 6.4e6  : x @ W_l
    float* base  = ws + (size_t)N_NODES * OUT_C;         // 6.4e6  : xWr -> out_base
    float* agg   = ws + (size_t)2 * N_NODES * OUT_C;     // 6.4e6  : agg -> att accum
    float* cnt   = ws + (size_t)3 * N_NODES * OUT_C;     // 1e5    : in-degree
    float* stats = cnt + N_NODES;                        // 128    : sum / sumsq

    // zero atomic accumulators each call (cnt+stats contiguous)
    hipMemsetAsync(agg, 0, (size_t)N_NODES * OUT_C * sizeof(float), stream);
    hipMemsetAsync(cnt, 0, (size_t)(N_NODES + 128) * sizeof(float), stream);

    const int node_tiles = N_NODES / 16;                 // 6250
    const int edge_tiles = N_EDGES / 16;                 // 62500

    node_gemm_kernel<<<(node_tiles + 7) / 8, 256, 0, stream>>>(x, W_l, W_r, h_l, base);
    scatter_agg_kernel<<<(N_EDGES * 64) / 256, 256, 0, stream>>>(ei, h_l, agg, cnt);
    node_finalize_kernel<<<(N_NODES * OUT_C) / 256, 256, 0, stream>>>(cnt, b_l, agg, base);
    edge_kernel<<<(edge_tiles + 7) / 8, 256, 0, stream>>>(ei, edge_attr, W_e, b_e,
                                                          W_att, b_att, base, agg);
    stats_kernel<<<240, 256, 0, stream>>>(base, agg, stats);
    final_kernel<<<(N_NODES * OUT_C) / 256, 256, 0, stream>>>(base, agg, stats,
                                                              gamma, beta, (float*)d_out);
}